// GATLayer_2001454760357
// MI455X (gfx1250) — compile-verified
//
#include <hip/hip_runtime.h>
#include <hip/hip_bf16.h>

#define N_NODES 50000
#define N_EDGES 800000
#define DIM 128

typedef __attribute__((ext_vector_type(2))) float v2f;
typedef __attribute__((ext_vector_type(8))) float v8f;

// ---------------------------------------------------------------------------
// Kernel 0: zero the CSR histogram + fill cursors
// ---------------------------------------------------------------------------
__global__ __launch_bounds__(256) void gat_init(int* __restrict__ counts,
                                                int* __restrict__ cursor) {
    int tid = blockIdx.x * blockDim.x + threadIdx.x;
    if (tid < N_NODES) {
        counts[tid] = 0;
        cursor[tid] = 0;
    }
}

// ---------------------------------------------------------------------------
// Kernel 1: z = h @ W^T via V_WMMA_F32_16X16X4_F32.
// One wave computes one 16x16 tile of z; K=128 in steps of 4.
// grid = 3125 M-tiles (exact: 50000/16), block = 256 (8 waves = 8 N-tiles).
// ---------------------------------------------------------------------------
__global__ __launch_bounds__(256) void gat_gemm(const float* __restrict__ h,
                                                const float* __restrict__ W,
                                                float* __restrict__ z) {
    const int lane = threadIdx.x & 31;
    const int wv   = threadIdx.x >> 5;       // 0..7 -> N tile
    const int m0   = blockIdx.x * 16;
    const int n0   = wv * 16;
    const int half = lane >> 4;              // 0: K0,K1 | 1: K2,K3
    const int lr   = lane & 15;

    const float* __restrict__ Arow = h + (size_t)(m0 + lr) * DIM; // A row m
    const float* __restrict__ Brow = W + (size_t)(n0 + lr) * DIM; // W row n = B col n

    v8f acc = {};
    #pragma unroll 4
    for (int k = 0; k < DIM; k += 4) {
        const int kk = k + half * 2;
        v2f a = *(const v2f*)(Arow + kk);
        v2f b = *(const v2f*)(Brow + kk);
        acc = __builtin_amdgcn_wmma_f32_16x16x4_f32(
            /*neg_a=*/false, a, /*neg_b=*/false, b,
            /*c_mod=*/(short)0, acc, /*reuse_a=*/false, /*reuse_b=*/false);
    }

    const int n = n0 + lr;
    #pragma unroll
    for (int v = 0; v < 8; ++v) {
        const int m = m0 + v + 8 * half;
        z[(size_t)m * DIM + n] = acc[v];
    }
}

// ---------------------------------------------------------------------------
// Kernel 2: s_src = z @ a_src, s_dst = z @ a_dst. One wave per node.
// ---------------------------------------------------------------------------
__global__ __launch_bounds__(256) void gat_gemv(const float* __restrict__ z,
                                                const float* __restrict__ a_attn,
                                                float* __restrict__ s_src,
                                                float* __restrict__ s_dst) {
    const int node = blockIdx.x * 8 + (threadIdx.x >> 5);
    const int lane = threadIdx.x & 31;
    if (node >= N_NODES) return;

    const float4 zv = ((const float4*)(z + (size_t)node * DIM))[lane];
    const float4 av = ((const float4*)(a_attn))[lane];          // a_src
    const float4 dv = ((const float4*)(a_attn + DIM))[lane];    // a_dst

    float ss = zv.x * av.x + zv.y * av.y + zv.z * av.z + zv.w * av.w;
    float sd = zv.x * dv.x + zv.y * dv.y + zv.z * dv.z + zv.w * dv.w;

    #pragma unroll
    for (int off = 16; off > 0; off >>= 1) {
        ss += __shfl_down(ss, off, 32);
        sd += __shfl_down(sd, off, 32);
    }
    if (lane == 0) {
        s_src[node] = ss;
        s_dst[node] = sd;
    }
}

// ---------------------------------------------------------------------------
// Kernel 3: histogram of destination degrees
// ---------------------------------------------------------------------------
__global__ __launch_bounds__(256) void gat_hist(const int* __restrict__ dst,
                                                int* __restrict__ counts) {
    int i = blockIdx.x * blockDim.x + threadIdx.x;
    if (i >= N_EDGES) return;
    atomicAdd(counts + dst[i], 1);
}

// ---------------------------------------------------------------------------
// Kernel 4: exclusive prefix scan over 50k degree counts (single workgroup,
// 1024-wide Hillis-Steele chunks with running carry). row_start[N_NODES]=total.
// ---------------------------------------------------------------------------
__global__ __launch_bounds__(1024) void gat_scan(const int* __restrict__ counts,
                                                 int* __restrict__ row_start) {
    __shared__ int lds[1024];
    __shared__ int carry_s;
    if (threadIdx.x == 0) carry_s = 0;
    __syncthreads();

    for (int base = 0; base < N_NODES; base += 1024) {
        const int idx = base + (int)threadIdx.x;
        const int v = (idx < N_NODES) ? counts[idx] : 0;
        int acc = v;
        lds[threadIdx.x] = acc;
        __syncthreads();
        #pragma unroll
        for (int off = 1; off < 1024; off <<= 1) {
            int t = (threadIdx.x >= (unsigned)off) ? lds[threadIdx.x - off] : 0;
            __syncthreads();
            acc += t;
            lds[threadIdx.x] = acc;
            __syncthreads();
        }
        const int c = carry_s;                       // last write synced above
        if (idx < N_NODES) row_start[idx] = c + acc - v;   // exclusive
        __syncthreads();
        if (threadIdx.x == 1023) carry_s = c + lds[1023];
        __syncthreads();
    }
    if (threadIdx.x == 0) row_start[N_NODES] = carry_s;
}

// ---------------------------------------------------------------------------
// Kernel 5: CSR fill. Slot = row_start[dst] + cursor++. Also computes the
// leaky-relu edge logit in the same pass (gathers s_src/s_dst once).
// ---------------------------------------------------------------------------
__global__ __launch_bounds__(256) void gat_fill(const int* __restrict__ src,
                                                const int* __restrict__ dst,
                                                const float* __restrict__ s_src,
                                                const float* __restrict__ s_dst,
                                                const int* __restrict__ row_start,
                                                int* __restrict__ cursor,
                                                int* __restrict__ csr_src,
                                                float* __restrict__ csr_e) {
    int i = blockIdx.x * blockDim.x + threadIdx.x;
    if (i >= N_EDGES) return;
    const int s = src[i];
    const int d = dst[i];
    const int pos = row_start[d] + atomicAdd(cursor + d, 1);
    const float v = s_src[s] + s_dst[d];
    csr_src[pos] = s;
    csr_e[pos]   = v > 0.0f ? v : 0.01f * v;     // jax leaky_relu slope = 0.01
}

// ---------------------------------------------------------------------------
// Kernel 6: per-destination-node fused softmax + aggregation + normalize.
// One wave per node; no atomics anywhere.
//   pass 1: segment max     (lane-strided, shfl_xor reduce)
//   pass 2: exp + sum       (rewrite csr_e with ex)
//   pass 3: acc += ex * z[src] ; 32 edges staged per float-load, shfl bcast
//   store : out = acc / sum  (0 for degree-0 nodes)
// ---------------------------------------------------------------------------
__global__ __launch_bounds__(256) void gat_node(const float* __restrict__ z,
                                                const int* __restrict__ row_start,
                                                const int* __restrict__ csr_src,
                                                float* __restrict__ csr_e,
                                                float* __restrict__ out) {
    const int node = blockIdx.x * 8 + (threadIdx.x >> 5);
    const int lane = threadIdx.x & 31;
    const int beg = row_start[node];
    const int end = row_start[node + 1];

    // ---- pass 1: max ----
    float mx = -INFINITY;
    for (int p = beg + lane; p < end; p += 32) mx = fmaxf(mx, csr_e[p]);
    #pragma unroll
    for (int off = 16; off > 0; off >>= 1) mx = fmaxf(mx, __shfl_xor(mx, off, 32));

    // ---- pass 2: exp + sum ----
    float sum = 0.0f;
    for (int p = beg + lane; p < end; p += 32) {
        const float ex = __expf(csr_e[p] - mx);
        csr_e[p] = ex;
        sum += ex;
    }
    #pragma unroll
    for (int off = 16; off > 0; off >>= 1) sum += __shfl_xor(sum, off, 32);

    // ---- pass 3: weighted gather-accumulate ----
    float4 acc = {0.0f, 0.0f, 0.0f, 0.0f};
    for (int p0 = beg; p0 < end; p0 += 32) {
        const int nn = min(32, end - p0);
        const int  pl = p0 + lane;
        const float wl = (pl < end) ? csr_e[pl]   : 0.0f;
        const int   sl = (pl < end) ? csr_src[pl] : 0;
        for (int j = 0; j < nn; ++j) {
            const float w = __shfl(wl, j, 32);
            const int   s = __shfl(sl, j, 32);
            const float4 zv = ((const float4*)(z + (size_t)s * DIM))[lane];
            acc.x += w * zv.x;
            acc.y += w * zv.y;
            acc.z += w * zv.z;
            acc.w += w * zv.w;
        }
    }

    const float inv = (end > beg) ? 1.0f / sum : 0.0f;
    float4 o = {acc.x * inv, acc.y * inv, acc.z * inv, acc.w * inv};
    ((float4*)(out + (size_t)node * DIM))[lane] = o;
}

// ---------------------------------------------------------------------------
extern "C" void kernel_launch(void* const* d_in, const int* in_sizes, int n_in,
                              void* d_out, int out_size, void* d_ws, size_t ws_size,
                              hipStream_t stream) {
    const float* h      = (const float*)d_in[0];   // 50000 x 128
    const float* W_fc   = (const float*)d_in[1];   // 128 x 128
    const float* a_attn = (const float*)d_in[2];   // 256
    const int*   src    = (const int*)d_in[3];     // 800000
    const int*   dst    = (const int*)d_in[4];     // 800000
    float*       out    = (float*)d_out;           // 50000 x 128
    (void)in_sizes; (void)n_in; (void)out_size; (void)ws_size;

    // workspace layout
    float* ws        = (float*)d_ws;
    float* z         = ws;                               // 6,400,000 f
    float* s_src     = z + (size_t)N_NODES * DIM;        // 50,000 f
    float* s_dst     = s_src + N_NODES;                  // 50,000 f
    float* csr_e     = s_dst + N_NODES;                  // 800,000 f
    int*   counts    = (int*)(csr_e + N_EDGES);          // 50,000 i
    int*   cursor    = counts + N_NODES;                 // 50,000 i
    int*   row_start = cursor + N_NODES;                 // 50,001 i
    int*   csr_src   = row_start + N_NODES + 1;          // 800,000 i

    // 0) zero histogram + cursors
    gat_init<<<(N_NODES + 255) / 256, 256, 0, stream>>>(counts, cursor);

    // 1) z = h @ W^T  (WMMA fp32)
    gat_gemm<<<N_NODES / 16, 256, 0, stream>>>(h, W_fc, z);

    // 2) attention score GEMVs
    gat_gemv<<<(N_NODES + 7) / 8, 256, 0, stream>>>(z, a_attn, s_src, s_dst);

    // 3) destination-degree histogram
    gat_hist<<<(N_EDGES + 255) / 256, 256, 0, stream>>>(dst, counts);

    // 4) exclusive scan -> row_start
    gat_scan<<<1, 1024, 0, stream>>>(counts, row_start);

    // 5) CSR fill + edge logits
    gat_fill<<<(N_EDGES + 255) / 256, 256, 0, stream>>>(src, dst, s_src, s_dst,
                                                        row_start, cursor,
                                                        csr_src, csr_e);

    // 6) fused per-node softmax + aggregation + normalize
    gat_node<<<N_NODES / 8, 256, 0, stream>>>(z, row_start, csr_src, csr_e, out);
}